// SpatialTransformer_74491912781868
// MI455X (gfx1250) — compile-verified
//
#include <hip/hip_runtime.h>
#include <stdint.h>

// Problem dims (fixed by the reference)
#define DD  160
#define HH  192
#define WW  160
#define DHW (DD * HH * WW)     // 4,915,200  (divisible by 256)
#define NB  2
#define BLK 256

#if defined(__has_builtin)
#if __has_builtin(__builtin_amdgcn_global_load_async_to_lds_b32)
#define USE_ASYNC_LDS 1
#endif
#endif

__global__ __launch_bounds__(BLK) void warp3d_trilinear(
    const float* __restrict__ src,   // [B,1,D,H,W]
    const float* __restrict__ flow,  // [B,3,D,H,W]
    float* __restrict__ out)         // [B,1,D,H,W]
{
    __shared__ float sflow[3 * BLK];

    const int tid = threadIdx.x;
    const int blk = blockIdx.x;
    const int blocks_per_batch = DHW / BLK;        // 19200
    const int b  = blk / blocks_per_batch;
    const int s0 = (blk - b * blocks_per_batch) * BLK;

    // Stage the 3 coalesced flow streams for this block into LDS.
    const float* fbase = flow + (size_t)b * 3 * DHW + s0 + tid;

#if USE_ASYNC_LDS
    {
        typedef __attribute__((address_space(1))) int g_i32;  // global
        typedef __attribute__((address_space(3))) int l_i32;  // LDS
#pragma unroll
        for (int c = 0; c < 3; ++c) {
            g_i32* g = (g_i32*)(uintptr_t)(fbase + c * DHW);
            // generic-pointer low 32 bits == LDS byte offset on gfx1250
            l_i32* l = (l_i32*)(uint32_t)(uintptr_t)&sflow[c * BLK + tid];
            __builtin_amdgcn_global_load_async_to_lds_b32(g, l, 0, 0);
        }
#if defined(__has_builtin) && __has_builtin(__builtin_amdgcn_s_wait_asynccnt)
        __builtin_amdgcn_s_wait_asynccnt(0);
#else
        asm volatile("s_wait_asynccnt 0" ::: "memory");
#endif
    }
    __syncthreads();
#else
#pragma unroll
    for (int c = 0; c < 3; ++c)
        sflow[c * BLK + tid] = fbase[c * DHW];
    __syncthreads();
#endif

    // Voxel coordinates (x-major flat index)
    const int s  = s0 + tid;
    const int x  = s % WW;
    const int t2 = s / WW;
    const int y  = t2 % HH;
    const int z  = t2 / HH;

    const float fz = sflow[tid];
    const float fy = sflow[BLK + tid];
    const float fx = sflow[2 * BLK + tid];

    // c_i = (p_i + flow_i) * s_i/(s_i-1) - 0.5   (align_corners=False unnormalize)
    const float cz = ((float)z + fz) * (float)(DD / (double)(DD - 1)) - 0.5f;
    const float cy = ((float)y + fy) * (float)(HH / (double)(HH - 1)) - 0.5f;
    const float cx = ((float)x + fx) * (float)(WW / (double)(WW - 1)) - 0.5f;

    const float zf = floorf(cz), yf = floorf(cy), xf = floorf(cx);
    const int   iz = (int)zf,    iy = (int)yf,    ix = (int)xf;
    const float wz1 = cz - zf,   wy1 = cy - yf,   wx1 = cx - xf;
    const float wz0 = 1.0f - wz1, wy0 = 1.0f - wy1, wx0 = 1.0f - wx1;

    const float* __restrict__ sb = src + (size_t)b * DHW;

    // 8 trilinear corners, zero padding: OOB corners contribute nothing.
    float acc = 0.0f;
#pragma unroll
    for (int dz = 0; dz < 2; ++dz) {
        const int zz = iz + dz;
        if ((unsigned)zz < (unsigned)DD) {
            const float wz = dz ? wz1 : wz0;
#pragma unroll
            for (int dy = 0; dy < 2; ++dy) {
                const int yy = iy + dy;
                if ((unsigned)yy < (unsigned)HH) {
                    const float wzy  = wz * (dy ? wy1 : wy0);
                    const size_t row = ((size_t)zz * HH + yy) * WW;
#pragma unroll
                    for (int dx = 0; dx < 2; ++dx) {
                        const int xx = ix + dx;
                        if ((unsigned)xx < (unsigned)WW) {
                            acc = fmaf(wzy * (dx ? wx1 : wx0), sb[row + xx], acc);
                        }
                    }
                }
            }
        }
    }

    // Streaming store: keep L2 free for src (src fits in the 192MB L2).
    __builtin_nontemporal_store(acc, out + (size_t)b * DHW + s);
}

extern "C" void kernel_launch(void* const* d_in, const int* in_sizes, int n_in,
                              void* d_out, int out_size, void* d_ws, size_t ws_size,
                              hipStream_t stream) {
    (void)in_sizes; (void)n_in; (void)out_size; (void)d_ws; (void)ws_size;
    const float* src  = (const float*)d_in[0];
    const float* flow = (const float*)d_in[1];
    float* out        = (float*)d_out;

    const int nblocks = (NB * DHW) / BLK;  // 38400 blocks of 256 threads
    warp3d_trilinear<<<nblocks, BLK, 0, stream>>>(src, flow, out);
}